// AGCRN_18219251270270
// MI455X (gfx1250) — compile-verified
//
#include <hip/hip_runtime.h>
#include <math.h>
#include <stdint.h>

// ---------------------------------------------------------------------------
// AGCRN (2-layer adaptive graph GRU) for MI455X / gfx1250.
// All matrix math via V_WMMA_F32_16X16X4_F32 (f32 exact, matches reference).
// The dominant A@v graph GEMM is LDS-staged with CDNA5 async global->LDS
// copies (ASYNCcnt) and fed to WMMA from LDS.
// Workspace layout (floats), total ~213M floats (~850 MB):
//   A(N*N) | Wg0n | Wu0n | Wg1n | Wu1n | bg0n | bu0n | bg1n | bu1n |
//   h0 | h1 | v0 | av0 | v1 | av1 | zr
// ---------------------------------------------------------------------------

#define BATCH   32
#define TSTEPS  24
#define NNODES  2048
#define EDIM    16
#define HID     64
#define C0REAL  65      // D_in + HID for layer 0
#define CP0     80      // padded to multiple of 16
#define CP1     128     // 2*HID, already multiple of 16
#define HORIZON 12
#define KT      64      // K-panel depth for the LDS-staged spmm

typedef __attribute__((ext_vector_type(2))) float v2f;
typedef __attribute__((ext_vector_type(8))) float v8f;

// ---------------- WMMA helpers (f32 16x16x4) --------------------------------

__device__ __forceinline__ v8f wmma4(v2f a, v2f b, v8f c) {
  // 8 args: (neg_a, A, neg_b, B, c_mod, C, reuse_a, reuse_b)
  return __builtin_amdgcn_wmma_f32_16x16x4_f32(false, a, false, b, (short)0, c,
                                               false, false);
}

// A-matrix 16x4 f32: lanes 0-15 hold M=lane, K={0,1}; lanes 16-31 K={2,3}.
__device__ __forceinline__ v2f load_a_frag(const float* p, int ld) {
  int lane = threadIdx.x & 31;
  int m = lane & 15;
  int k = (lane >> 4) << 1;   // 0 or 2
  v2f a;
  a.x = p[m * ld + k];
  a.y = p[m * ld + k + 1];
  return a;
}

// B-matrix 4x16 f32: lanes 0-15 hold N=lane, K={0,1}; lanes 16-31 K={2,3}.
__device__ __forceinline__ v2f load_b_frag(const float* p, int ld) {
  int lane = threadIdx.x & 31;
  int nn = lane & 15;
  int k = (lane >> 4) << 1;
  v2f b;
  b.x = p[k * ld + nn];
  b.y = p[(k + 1) * ld + nn];
  return b;
}

// C/D 16x16 f32: VGPR i -> row (i + 8*(lane>=16)), col = lane&15.
__device__ __forceinline__ void store_d_frag(float* p, int ld, v8f d) {
  int lane = threadIdx.x & 31;
  int col = lane & 15;
  int rbase = (lane >> 4) << 3;
#pragma unroll
  for (int i = 0; i < 8; ++i) p[(rbase + i) * ld + col] = d[i];
}

// ---------------- CDNA5 async global->LDS copy ------------------------------
// GLOBAL_LOAD_ASYNC_TO_LDS_B128, GVS mode: SGPR 64-bit base + VGPR byte
// offset; VDST VGPR holds the LDS byte address. Tracked by ASYNCcnt.
// Generic pointers to __shared__ map to LDS with the offset in the low 32
// bits (flat-LDS aperture rule), so truncation yields the ds address.

__device__ __forceinline__ uint32_t lds_addr(const void* p) {
  return (uint32_t)(uintptr_t)p;
}

__device__ __forceinline__ void async_copy_b128(void* lds_dst,
                                                const float* gbase,
                                                uint32_t byte_off) {
  asm volatile("global_load_async_to_lds_b128 %0, %1, %2"
               :
               : "v"(lds_addr(lds_dst)), "v"(byte_off), "s"(gbase)
               : "memory");
}

__device__ __forceinline__ void wait_asynccnt0() {
  asm volatile("s_wait_asynccnt 0x0" ::: "memory");
}

// ---------------- misc kernels ----------------------------------------------

__global__ void zero_kernel(float* p, long total) {
  long i = (long)blockIdx.x * blockDim.x + threadIdx.x;
  if (i < total) p[i] = 0.0f;
}

// A = softmax(relu(E E^T), axis=1). One 256-thread block per row.
__global__ void supports_kernel(const float* __restrict__ E,
                                float* __restrict__ A, int N) {
  __shared__ float red[256];
  __shared__ float erow[EDIM];
  int n = blockIdx.x;
  int tid = threadIdx.x;
  if (tid < EDIM) erow[tid] = E[n * EDIM + tid];
  __syncthreads();

  float vals[NNODES / 256];
  float mx = 0.0f;  // relu() >= 0
#pragma unroll
  for (int i = 0; i < NNODES / 256; ++i) {
    int m = tid + i * 256;
    float d = 0.0f;
#pragma unroll
    for (int e = 0; e < EDIM; ++e) d += erow[e] * E[m * EDIM + e];
    d = fmaxf(d, 0.0f);
    vals[i] = d;
    mx = fmaxf(mx, d);
  }
  red[tid] = mx;
  __syncthreads();
  for (int s = 128; s > 0; s >>= 1) {
    if (tid < s) red[tid] = fmaxf(red[tid], red[tid + s]);
    __syncthreads();
  }
  float rmax = red[0];
  __syncthreads();
  float sum = 0.0f;
#pragma unroll
  for (int i = 0; i < NNODES / 256; ++i) {
    vals[i] = __expf(vals[i] - rmax);
    sum += vals[i];
  }
  red[tid] = sum;
  __syncthreads();
  for (int s = 128; s > 0; s >>= 1) {
    if (tid < s) red[tid] += red[tid + s];
    __syncthreads();
  }
  float inv = 1.0f / red[0];
#pragma unroll
  for (int i = 0; i < NNODES / 256; ++i)
    A[(long)n * N + tid + i * 256] = vals[i] * inv;
}

// Wn[n][k][cpad][o] = sum_e E[n,e] * W[e][k][c][o]  (0 for padded c).
// GEMM M=n(16) N=16 cols of flattened (k,cpad,o), K=e=16 (4 wmma steps).
__global__ void node_weights_kernel(const float* __restrict__ E,
                                    const float* __restrict__ W,
                                    float* __restrict__ Wn,
                                    int Creal, int Cpad, int O) {
  int n0 = blockIdx.x * 16;
  int j0 = blockIdx.y * 16;
  int kc = j0 / O;          // k*Cpad + c
  int o0 = j0 - kc * O;
  int k = kc / Cpad;
  int c = kc - k * Cpad;
  v8f acc = {};
  if (c < Creal) {
    const float* Wp = W + ((long)(k * Creal + c)) * O + o0;  // + e*estride
    int estride = 2 * Creal * O;  // CHEB_K == 2
#pragma unroll
    for (int e0 = 0; e0 < EDIM; e0 += 4) {
      v2f a = load_a_frag(E + n0 * EDIM + e0, EDIM);
      v2f b = load_b_frag(Wp + (long)e0 * estride, estride);
      acc = wmma4(a, b, acc);
    }
  }
  int stride = 2 * Cpad * O;
  store_d_frag(Wn + (long)n0 * stride + j0, stride, acc);
}

// bn[n][o] = sum_e E[n,e] * b[e][o]
__global__ void node_bias_kernel(const float* __restrict__ E,
                                 const float* __restrict__ b,
                                 float* __restrict__ bn, int O) {
  int idx = blockIdx.x * blockDim.x + threadIdx.x;
  int o = idx % O;
  int n = idx / O;
  float s = 0.0f;
#pragma unroll
  for (int e = 0; e < EDIM; ++e) s += E[n * EDIM + e] * b[e * O + o];
  bn[idx] = s;
}

// v[b][n][c] = concat(input_t, (z?)*h)[c], zero pad up to Cpad.
__global__ void build_v_kernel(float* __restrict__ v,
                               const float* __restrict__ in0, long in_bstride,
                               int in_nstride, int Cin,
                               const float* __restrict__ h,
                               const float* __restrict__ zr,  // null => plain h
                               int Cpad, int N, long total) {
  long idx = (long)blockIdx.x * blockDim.x + threadIdx.x;
  if (idx >= total) return;
  int c = (int)(idx % Cpad);
  long bn = idx / Cpad;
  int n = (int)(bn % N);
  int b = (int)(bn / N);
  float val = 0.0f;
  if (c < Cin) {
    val = in0[(long)b * in_bstride + (long)n * in_nstride + c];
  } else if (c < Cin + HID) {
    int cc = c - Cin;
    float hv = h[((long)b * N + n) * HID + cc];
    if (zr) hv *= zr[((long)b * N + n) * (2 * HID) + cc];  // z = first half
    val = hv;
  }
  v[idx] = val;
}

// av[b][n][c] = sum_m A[n][m] * v[b][m][c].
// Block = 8 waves computing a 16(n) x Cpad(c) tile for one batch element.
// Per K-step of KT=64: async-copy A panel (16xKT) and v panel (KT x Cpad)
// to LDS (global_load_async_to_lds_b128 / s_wait_asynccnt), then each wave
// runs KT/4 WMMA steps from LDS for its 16-column tile.
// grid = (N/16, B), blockDim = 256.
__global__ void __launch_bounds__(256)
spmm_kernel(const float* __restrict__ A, const float* __restrict__ v,
            float* __restrict__ av, int N, int Cpad) {
  __shared__ __attribute__((aligned(16))) float sA[16 * KT];
  __shared__ __attribute__((aligned(16))) float sV[KT * CP1];

  int tid = threadIdx.x;
  int wave = tid >> 5;
  int n0 = blockIdx.x * 16;
  int b = blockIdx.y;
  const float* vb = v + (long)b * N * Cpad;
  const float* Ap = A + (long)n0 * N;

  int nctiles = Cpad / 16;       // 5 (CP0) or 8 (CP1)
  int c0 = wave * 16;            // this wave's c-tile (may be inactive)
  int vloads = Cpad / 16;        // per-thread b128 copies of the v panel
  int vld4 = Cpad / 4;           // b128 segments per v row

  v8f acc = {};
  for (int kk = 0; kk < N; kk += KT) {
    // ---- stage A panel: 16 rows x KT, one b128 per thread -----------------
    {
      int row = tid >> 4;            // 0..15
      int seg = tid & 15;            // 0..15 -> 4 floats
      uint32_t goff = (uint32_t)(((row * N) + kk + seg * 4) * 4);
      async_copy_b128(&sA[row * KT + seg * 4], Ap, goff);
    }
    // ---- stage v panel: KT rows x Cpad, Cpad/16 b128 per thread -----------
    for (int r = 0; r < vloads; ++r) {
      int i = r * 256 + tid;
      int row = i / vld4;
      int seg = i - row * vld4;
      uint32_t goff = (uint32_t)((((kk + row) * Cpad) + seg * 4) * 4);
      async_copy_b128(&sV[row * Cpad + seg * 4], vb, goff);
    }
    wait_asynccnt0();
    __syncthreads();

    if (wave < nctiles) {           // wave-uniform: EXEC all-ones inside
#pragma unroll
      for (int k4 = 0; k4 < KT; k4 += 4) {
        v2f a = load_a_frag(&sA[k4], KT);
        v2f bfr = load_b_frag(&sV[k4 * Cpad + c0], Cpad);
        acc = wmma4(a, bfr, acc);
      }
    }
    __syncthreads();
  }
  if (wave < nctiles)
    store_d_frag(av + ((long)b * N + n0) * Cpad + c0, Cpad, acc);
}

// Per-node GRU GEMM:  acc[b][o] = sum_{kc} vg[b][kc] * Wn[n][kc][o]
// vg: kc < Cpad -> v,  else -> av.   mode 0: zr = sigmoid(acc + bias)
// mode 1: hc = tanh(acc + bias); h = r*h + (1-r)*hc  (r = zr[..., 64+o])
// grid = (N, O/16, B/16), one wave per block.
__global__ void gru_gemm_kernel(const float* __restrict__ v,
                                const float* __restrict__ av,
                                const float* __restrict__ Wn,
                                const float* __restrict__ bn,
                                float* __restrict__ zr, float* __restrict__ h,
                                int N, int Cpad, int O, int mode) {
  int n = blockIdx.x;
  int o0 = blockIdx.y * 16;
  int b0 = blockIdx.z * 16;
  int K = 2 * Cpad;
  const float* Wp = Wn + (long)n * K * O + o0;
  v8f acc = {};
  for (int kk = 0; kk < K; kk += 4) {
    const float* src = (kk < Cpad) ? v : av;
    int c0 = (kk < Cpad) ? kk : kk - Cpad;
    __builtin_prefetch(Wp + (long)(kk + 16) * O, 0, 0);  // global_prefetch_b8
    v2f a = load_a_frag(src + ((long)b0 * N + n) * Cpad + c0, N * Cpad);
    v2f bfr = load_b_frag(Wp + (long)kk * O, O);
    acc = wmma4(a, bfr, acc);
  }
  int lane = threadIdx.x & 31;
  int col = lane & 15;
  int rbase = (lane >> 4) << 3;
  int o = o0 + col;
  float bias = bn[n * O + o];
#pragma unroll
  for (int i = 0; i < 8; ++i) {
    int b = b0 + rbase + i;
    float val = acc[i] + bias;
    if (mode == 0) {
      zr[((long)b * N + n) * (2 * HID) + o] = 1.0f / (1.0f + __expf(-val));
    } else {
      float r = zr[((long)b * N + n) * (2 * HID) + HID + o];
      float hc = tanhf(val);
      long hi = ((long)b * N + n) * HID + o;
      h[hi] = r * h[hi] + (1.0f - r) * hc;
    }
  }
}

// out[b][hor][n][0] = sum_k h1[b][n][k] * Wc[hor][k] + bc[hor]
__global__ void proj_kernel(const float* __restrict__ h1,
                            const float* __restrict__ Wc,
                            const float* __restrict__ bc,
                            float* __restrict__ out, int N) {
  long idx = (long)blockIdx.x * blockDim.x + threadIdx.x;
  int n = (int)(idx % N);
  long bo = idx / N;
  int o = (int)(bo % HORIZON);
  int b = (int)(bo / HORIZON);
  float s = bc[o];
#pragma unroll
  for (int k = 0; k < HID; ++k)
    s += h1[((long)b * N + n) * HID + k] * Wc[o * HID + k];
  out[idx] = s;
}

// ---------------------------------------------------------------------------

extern "C" void kernel_launch(void* const* d_in, const int* in_sizes, int n_in,
                              void* d_out, int out_size, void* d_ws,
                              size_t ws_size, hipStream_t stream) {
  const float* x   = (const float*)d_in[0];
  const float* Emb = (const float*)d_in[1];
  const float* Wg0 = (const float*)d_in[2];
  const float* bg0 = (const float*)d_in[3];
  const float* Wu0 = (const float*)d_in[4];
  const float* bu0 = (const float*)d_in[5];
  const float* Wg1 = (const float*)d_in[6];
  const float* bg1 = (const float*)d_in[7];
  const float* Wu1 = (const float*)d_in[8];
  const float* bu1 = (const float*)d_in[9];
  const float* Wc  = (const float*)d_in[10];
  const float* bc  = (const float*)d_in[11];
  float* out = (float*)d_out;

  const int N = NNODES, B = BATCH;
  float* ws = (float*)d_ws;
  size_t off = 0;
  auto take = [&](size_t n) { float* p = ws + off; off += n; return p; };

  float* A    = take((size_t)N * N);
  float* Wg0n = take((size_t)N * 2 * CP0 * 128);
  float* Wu0n = take((size_t)N * 2 * CP0 * 64);
  float* Wg1n = take((size_t)N * 2 * CP1 * 128);
  float* Wu1n = take((size_t)N * 2 * CP1 * 64);
  float* bg0n = take((size_t)N * 128);
  float* bu0n = take((size_t)N * 64);
  float* bg1n = take((size_t)N * 128);
  float* bu1n = take((size_t)N * 64);
  float* h0   = take((size_t)B * N * HID);
  float* h1   = take((size_t)B * N * HID);
  float* v0   = take((size_t)B * N * CP0);
  float* av0  = take((size_t)B * N * CP0);
  float* v1   = take((size_t)B * N * CP1);
  float* av1  = take((size_t)B * N * CP1);
  float* zr   = take((size_t)B * N * 2 * HID);

  // --- setup: adjacency, per-node weights/biases, zero hidden state --------
  supports_kernel<<<N, 256, 0, stream>>>(Emb, A, N);

  node_weights_kernel<<<dim3(N / 16, 2 * CP0 * 128 / 16), 32, 0, stream>>>(
      Emb, Wg0, Wg0n, C0REAL, CP0, 128);
  node_weights_kernel<<<dim3(N / 16, 2 * CP0 * 64 / 16), 32, 0, stream>>>(
      Emb, Wu0, Wu0n, C0REAL, CP0, 64);
  node_weights_kernel<<<dim3(N / 16, 2 * CP1 * 128 / 16), 32, 0, stream>>>(
      Emb, Wg1, Wg1n, CP1, CP1, 128);
  node_weights_kernel<<<dim3(N / 16, 2 * CP1 * 64 / 16), 32, 0, stream>>>(
      Emb, Wu1, Wu1n, CP1, CP1, 64);

  node_bias_kernel<<<N * 128 / 256, 256, 0, stream>>>(Emb, bg0, bg0n, 128);
  node_bias_kernel<<<N * 64 / 256, 256, 0, stream>>>(Emb, bu0, bu0n, 64);
  node_bias_kernel<<<N * 128 / 256, 256, 0, stream>>>(Emb, bg1, bg1n, 128);
  node_bias_kernel<<<N * 64 / 256, 256, 0, stream>>>(Emb, bu1, bu1n, 64);

  long hsz = (long)B * N * HID;
  zero_kernel<<<(hsz + 255) / 256, 256, 0, stream>>>(h0, hsz);
  zero_kernel<<<(hsz + 255) / 256, 256, 0, stream>>>(h1, hsz);

  long tot0 = (long)B * N * CP0;
  long tot1 = (long)B * N * CP1;

  // --- fused timestep loop: layer0 step then layer1 step -------------------
  for (int t = 0; t < TSTEPS; ++t) {
    const float* xt = x + (long)t * N;  // x[b][t][n][0], b-stride = T*N

    // ---- layer 0 gate ----
    build_v_kernel<<<(tot0 + 255) / 256, 256, 0, stream>>>(
        v0, xt, (long)TSTEPS * N, 1, 1, h0, nullptr, CP0, N, tot0);
    spmm_kernel<<<dim3(N / 16, B), 256, 0, stream>>>(A, v0, av0, N, CP0);
    gru_gemm_kernel<<<dim3(N, 128 / 16, B / 16), 32, 0, stream>>>(
        v0, av0, Wg0n, bg0n, zr, nullptr, N, CP0, 128, 0);
    // ---- layer 0 candidate + h update ----
    build_v_kernel<<<(tot0 + 255) / 256, 256, 0, stream>>>(
        v0, xt, (long)TSTEPS * N, 1, 1, h0, zr, CP0, N, tot0);
    spmm_kernel<<<dim3(N / 16, B), 256, 0, stream>>>(A, v0, av0, N, CP0);
    gru_gemm_kernel<<<dim3(N, 64 / 16, B / 16), 32, 0, stream>>>(
        v0, av0, Wu0n, bu0n, zr, h0, N, CP0, 64, 1);

    // ---- layer 1 gate (input = freshly updated h0) ----
    build_v_kernel<<<(tot1 + 255) / 256, 256, 0, stream>>>(
        v1, h0, (long)N * HID, HID, HID, h1, nullptr, CP1, N, tot1);
    spmm_kernel<<<dim3(N / 16, B), 256, 0, stream>>>(A, v1, av1, N, CP1);
    gru_gemm_kernel<<<dim3(N, 128 / 16, B / 16), 32, 0, stream>>>(
        v1, av1, Wg1n, bg1n, zr, nullptr, N, CP1, 128, 0);
    // ---- layer 1 candidate + h update ----
    build_v_kernel<<<(tot1 + 255) / 256, 256, 0, stream>>>(
        v1, h0, (long)N * HID, HID, HID, h1, zr, CP1, N, tot1);
    spmm_kernel<<<dim3(N / 16, B), 256, 0, stream>>>(A, v1, av1, N, CP1);
    gru_gemm_kernel<<<dim3(N, 64 / 16, B / 16), 32, 0, stream>>>(
        v1, av1, Wu1n, bu1n, zr, h1, N, CP1, 64, 1);
  }

  // --- projection ----------------------------------------------------------
  long osz = (long)B * HORIZON * N;
  proj_kernel<<<(osz + 255) / 256, 256, 0, stream>>>(h1, Wc, bc, out, N);
}